// GATEv2Conv_72164040507948
// MI455X (gfx1250) — compile-verified
//
#include <hip/hip_runtime.h>

typedef float v2f __attribute__((ext_vector_type(2)));
typedef float v8f __attribute__((ext_vector_type(8)));

#define NEG_SLOPE 0.01f

__device__ __forceinline__ float lrelu(float v) { return v > 0.0f ? v : NEG_SLOPE * v; }

// float atomic max via integer atomics (works for all finite values, init = -inf)
__device__ __forceinline__ void atomicMaxF32(float* addr, float val) {
    if (val >= 0.0f) atomicMax((int*)addr, __float_as_int(val));
    else             atomicMin((unsigned int*)addr, __float_as_uint(val));
}

// ---------------------------------------------------------------------------
// K1: per-node. xt = x @ W2^T, xa = x @ W1[:, :64]^T, r[n] = sum lrelu(x)*attn2
//     Also initializes m=-inf, denom=0, hraw(d_out)=0.
// One wave handles a 16-row strip. f32 WMMA 16x16x4, K=64 -> 16 k-steps.
// ---------------------------------------------------------------------------
__global__ void k1_node(const float* __restrict__ x, const float* __restrict__ W1,
                        const float* __restrict__ W2, const float* __restrict__ attn,
                        float* __restrict__ xa, float* __restrict__ xt,
                        float* __restrict__ rr, float* __restrict__ mm,
                        float* __restrict__ dn, float* __restrict__ hraw, int N)
{
    const int lane = threadIdx.x & 31;
    const int wave = (int)((blockIdx.x * blockDim.x + threadIdx.x) >> 5);
    const int r0 = wave * 16;
    if (r0 >= N) return;                       // wave-uniform
    const int lo = lane & 15;
    const int hi = lane >> 4;                  // 0 | 1
    const int rowA  = r0 + lo;
    const int rowAc = rowA < N ? rowA : N - 1; // clamped load row (EXEC must stay full)

    v8f acc2[4] = {};   // x @ W2^T   (4 N-tiles of 16)
    v8f acc1[4] = {};   // x @ W1a^T
    float rpart = 0.0f;

#pragma unroll
    for (int ks = 0; ks < 16; ++ks) {
        const int c = ks * 4 + 2 * hi;         // this lane's K columns: c, c+1
        v2f a = *(const v2f*)(x + (size_t)rowAc * 64 + c);
        rpart += lrelu(a.x) * attn[64 + c] + lrelu(a.y) * attn[64 + c + 1];
#pragma unroll
        for (int t = 0; t < 4; ++t) {
            v2f b2 = *(const v2f*)(W2 + (size_t)(16 * t + lo) * 64 + c);
            acc2[t] = __builtin_amdgcn_wmma_f32_16x16x4_f32(
                false, a, false, b2, (short)0, acc2[t], false, false);
            v2f b1 = *(const v2f*)(W1 + (size_t)(16 * t + lo) * 80 + c);
            acc1[t] = __builtin_amdgcn_wmma_f32_16x16x4_f32(
                false, a, false, b1, (short)0, acc1[t], false, false);
        }
    }

    // r[row] = rpart(lane) + rpart(lane+16)
    rpart += __shfl_down(rpart, 16);
    if (hi == 0 && rowA < N) {
        rr[rowA] = rpart;
        mm[rowA] = -__builtin_inff();
        dn[rowA] = 0.0f;
    }

    // store C tiles: VGPR j holds row (r0+j) in lanes 0-15, row (r0+j+8) in lanes 16-31
#pragma unroll
    for (int t = 0; t < 4; ++t) {
#pragma unroll
        for (int j = 0; j < 8; ++j) {
            const int row = r0 + j + 8 * hi;
            if (row < N) {
                xt[(size_t)row * 64 + 16 * t + lo] = acc2[t][j];
                xa[(size_t)row * 64 + 16 * t + lo] = acc1[t][j];
            }
        }
    }

    // zero the aggregation buffer (d_out) for this strip
#pragma unroll
    for (int i = 0; i < 32; ++i) {
        const int idx = r0 * 64 + i * 32 + lane;
        if (idx < N * 64) hraw[idx] = 0.0f;
    }
}

// ---------------------------------------------------------------------------
// K2: per-edge logits.  f = xa[src] + edge_attr @ W1b^T (K=16 WMMA),
//     logit = dot(lrelu(f), attn1) + r[dst];  atomic seg-max into m[dst].
// One wave handles 16 edges.
// ---------------------------------------------------------------------------
__global__ void k2_edge(const float* __restrict__ edge_attr, const int* __restrict__ src,
                        const int* __restrict__ dst, const float* __restrict__ W1,
                        const float* __restrict__ attn, const float* __restrict__ xa,
                        const float* __restrict__ rr, float* __restrict__ mm,
                        float* __restrict__ lg, int E)
{
    const int lane = threadIdx.x & 31;
    const int wave = (int)((blockIdx.x * blockDim.x + threadIdx.x) >> 5);
    const int e0 = wave * 16;
    if (e0 >= E) return;                       // wave-uniform
    const int lo = lane & 15;
    const int hi = lane >> 4;
    const int eA = min(e0 + lo, E - 1);
    const int sidx = src[eA];
    const int didx = dst[eA];

    v8f acc[4] = {};
#pragma unroll
    for (int ks = 0; ks < 4; ++ks) {
        const int c = ks * 4 + 2 * hi;
        v2f a = *(const v2f*)(edge_attr + (size_t)eA * 16 + c);
#pragma unroll
        for (int t = 0; t < 4; ++t) {
            v2f b = *(const v2f*)(W1 + (size_t)(16 * t + lo) * 80 + 64 + c);
            acc[t] = __builtin_amdgcn_wmma_f32_16x16x4_f32(
                false, a, false, b, (short)0, acc[t], false, false);
        }
    }

    float av[4];
#pragma unroll
    for (int t = 0; t < 4; ++t) av[t] = attn[16 * t + lo];

    float lgj[8];
    int   dj[8];
#pragma unroll
    for (int j = 0; j < 8; ++j) {
        const int sl = j + 8 * hi;             // edge j (lo half) / j+8 (hi half)
        const int srcj = __shfl(sidx, sl);
        dj[j]          = __shfl(didx, sl);
        float pj = 0.0f;
#pragma unroll
        for (int t = 0; t < 4; ++t) {
            float f = acc[t][j] + xa[(size_t)srcj * 64 + 16 * t + lo];
            pj += lrelu(f) * av[t];
        }
        pj += __shfl_xor(pj, 1);
        pj += __shfl_xor(pj, 2);
        pj += __shfl_xor(pj, 4);
        pj += __shfl_xor(pj, 8);               // 16-lane group sum
        lgj[j] = pj;
    }

    if (lo == 0) {                             // lane 0 -> edges e0..e0+7, lane 16 -> e0+8..15
#pragma unroll
        for (int j = 0; j < 8; ++j) {
            const int e = e0 + j + 8 * hi;
            if (e < E) {
                const float logit = lgj[j] + rr[dj[j]];
                lg[e] = logit;
                atomicMaxF32(&mm[dj[j]], logit);
            }
        }
    }
}

// ---------------------------------------------------------------------------
// K3: p = exp(logit - m[dst]); denom[dst] += p; hraw[dst,:] += p * xt[src,:]
// One wave per edge, 2 columns per lane (coalesced 256B row).
// ---------------------------------------------------------------------------
__global__ void k3_agg(const int* __restrict__ src, const int* __restrict__ dst,
                       const float* __restrict__ lg, const float* __restrict__ mm,
                       const float* __restrict__ xt, float* __restrict__ dn,
                       float* __restrict__ hraw, int E)
{
    const int lane = threadIdx.x & 31;
    const int e = (int)((blockIdx.x * blockDim.x + threadIdx.x) >> 5);
    if (e >= E) return;
    const int d = dst[e];
    const int s = src[e];
    const float p = __expf(lg[e] - mm[d]);
    if (lane == 0) atomicAdd(&dn[d], p);
    v2f v = *(const v2f*)(xt + (size_t)s * 64 + 2 * lane);
    atomicAdd(&hraw[(size_t)d * 64 + 2 * lane],     p * v.x);
    atomicAdd(&hraw[(size_t)d * 64 + 2 * lane + 1], p * v.y);
}

// ---------------------------------------------------------------------------
// K4: h = hraw / denom + bias   (guard empty segments -> bias)
// ---------------------------------------------------------------------------
__global__ void k4_final(const float* __restrict__ dn, const float* __restrict__ bias,
                         float* __restrict__ h, int total)
{
    const int i = blockIdx.x * blockDim.x + threadIdx.x;
    if (i >= total) return;
    const float d = dn[i >> 6];
    const float v = h[i];
    h[i] = (d > 0.0f ? v / d : 0.0f) + bias[i & 63];
}

extern "C" void kernel_launch(void* const* d_in, const int* in_sizes, int n_in,
                              void* d_out, int out_size, void* d_ws, size_t ws_size,
                              hipStream_t stream)
{
    const float* x         = (const float*)d_in[0];
    const float* edge_attr = (const float*)d_in[1];
    const int*   src       = (const int*)d_in[2];
    const int*   dst       = (const int*)d_in[3];
    const float* W1        = (const float*)d_in[4];   // [64, 80]
    const float* W2        = (const float*)d_in[5];   // [64, 64]
    const float* attn      = (const float*)d_in[6];   // [128]
    const float* bias      = (const float*)d_in[7];   // [64]

    const int N = in_sizes[0] / 64;
    const int E = in_sizes[2];

    float* ws = (float*)d_ws;
    float* xa = ws;                              // [N,64]
    float* xt = xa + (size_t)N * 64;             // [N,64]
    float* rr = xt + (size_t)N * 64;             // [N]
    float* mm = rr + N;                          // [N]
    float* dn = mm + N;                          // [N]
    float* lg = dn + N;                          // [E]
    float* h  = (float*)d_out;                   // [N,64] (also aggregation buffer)

    const dim3 blk(256);
    const int strips = (N + 15) / 16;
    const int etiles = (E + 15) / 16;

    k1_node<<<(strips + 7) / 8, blk, 0, stream>>>(x, W1, W2, attn, xa, xt, rr, mm, dn, h, N);
    k2_edge<<<(etiles + 7) / 8, blk, 0, stream>>>(edge_attr, src, dst, W1, attn, xa, rr, mm, lg, E);
    k3_agg<<<(E + 7) / 8, blk, 0, stream>>>(src, dst, lg, mm, xt, dn, h, E);
    k4_final<<<((size_t)N * 64 + 255) / 256, blk, 0, stream>>>(dn, bias, h, N * 64);
}